// GCN_81063212744814
// MI455X (gfx1250) — compile-verified
//
#include <hip/hip_runtime.h>
#include <hip/hip_bf16.h>

typedef __attribute__((ext_vector_type(2))) float v2f;
typedef __attribute__((ext_vector_type(8))) float v8f;

#define IN_CH 128      // = HID_CH = 128 for this problem
#define MTILES 4       // 16-row M-tiles per block (64-row strip)

// ---------------------------------------------------------------------------
// Degree / normalization kernels
// ---------------------------------------------------------------------------
__global__ void gcn_deg_init(float* __restrict__ deg, int N) {
    int i = blockIdx.x * blockDim.x + threadIdx.x;
    if (i < N) deg[i] = 1.0f;            // self-loop contributes 1
}

__global__ void gcn_deg_accum(const int* __restrict__ dst, float* __restrict__ deg, int E) {
    int stride = gridDim.x * blockDim.x;
    for (int e = blockIdx.x * blockDim.x + threadIdx.x; e < E; e += stride)
        atomicAdd(&deg[dst[e]], 1.0f);
}

__global__ void gcn_deg_rsqrt(float* __restrict__ deg, int N) {
    int i = blockIdx.x * blockDim.x + threadIdx.x;
    if (i < N) deg[i] = rsqrtf(deg[i]);  // deg >= 1 always (self loops)
}

// ---------------------------------------------------------------------------
// GEMM: Out[M x 128] = A[M x 128] @ W[128 x 128] via V_WMMA_F32_16X16X4_F32.
// Block = 256 threads = 8 wave32s. Wave w owns N-tile n0 = 16*w for a 64-row
// strip. B fragments (all 32 K-steps) are hoisted into 64 VGPRs and reused
// across 4 M-tiles; A strip (64x128 = 32KB) staged in LDS. Inner loop is
// ds_load_2addr_b64 + v_wmma only.
// ---------------------------------------------------------------------------
__global__ __launch_bounds__(256) void gcn_gemm_wmma(
    const float* __restrict__ A, const float* __restrict__ W,
    float* __restrict__ Out, int M)
{
    __shared__ float As[MTILES * 16 * IN_CH];   // 32 KB

    const int tid  = threadIdx.x;
    const int wave = tid >> 5;
    const int lane = tid & 31;
    const int m0   = blockIdx.x * (MTILES * 16);
    if (m0 >= M) return;                        // uniform across block

    const int n0   = wave * 16;                 // this wave's N tile
    const int mrow = lane & 15;                 // A row within tile
    const int ncol = lane & 15;                 // B/D column
    const int koff = (lane >> 4) << 1;          // lanes 0-15 -> K+0/1, 16-31 -> K+2/3

    // Hoist all B fragments for this N-tile into registers (64 VGPRs).
    v2f bfr[32];
    #pragma unroll
    for (int ks = 0; ks < 32; ++ks) {
        const int k = ks * 4;
        bfr[ks].x = W[(size_t)(k + koff)     * IN_CH + n0 + ncol];
        bfr[ks].y = W[(size_t)(k + koff + 1) * IN_CH + n0 + ncol];
    }

    // Cooperative load of the 64x128 A strip (8192 floats = 2048 float4),
    // zero-filling rows past M (tail block).
    {
        const float4* srcp = (const float4*)(A + (size_t)m0 * IN_CH);
        float4*       dstp = (float4*)As;
        #pragma unroll
        for (int i = 0; i < 8; ++i) {
            const int idx = tid + i * 256;      // 0..2047
            const int row = idx >> 5;           // 32 float4 per 128-col row
            float4 v;
            if (m0 + row < M) v = srcp[idx];
            else              v = make_float4(0.f, 0.f, 0.f, 0.f);
            dstp[idx] = v;
        }
    }
    __syncthreads();

    #pragma unroll
    for (int mt = 0; mt < MTILES; ++mt) {
        const float* arow = As + (size_t)(mt * 16 + mrow) * IN_CH;
        v8f acc = {};
        #pragma unroll
        for (int ks = 0; ks < 32; ++ks) {
            v2f a;
            a.x = arow[ks * 4 + koff];
            a.y = arow[ks * 4 + koff + 1];
            // 8 args: (neg_a, A, neg_b, B, c_mod, C, reuse_a, reuse_b)
            acc = __builtin_amdgcn_wmma_f32_16x16x4_f32(
                false, a, false, bfr[ks], (short)0, acc, false, false);
        }
        // D 16x16: lane holds rows mbase..mbase+7 at column ncol
        const int m     = m0 + mt * 16;
        const int mbase = (lane >> 4) << 3;
        float* outp = Out + (size_t)(m + mbase) * IN_CH + n0 + ncol;
        #pragma unroll
        for (int j = 0; j < 8; ++j)
            if (m + mbase + j < M)
                outp[(size_t)j * IN_CH] = acc[j];
    }
}

// ---------------------------------------------------------------------------
// agg[i] = hW[i] * dis[i]^2   (self-loop term initializes the accumulator)
// ---------------------------------------------------------------------------
__global__ void gcn_agg_init(const float* __restrict__ hw, const float* __restrict__ dis,
                             float* __restrict__ agg, int total) {
    int idx = blockIdx.x * blockDim.x + threadIdx.x;
    if (idx < total) {
        float d = dis[idx >> 7];
        agg[idx] = hw[idx] * d * d;
    }
}

// ---------------------------------------------------------------------------
// Wave-per-edge scatter: agg[dst] += hW[src] * dis[src]*dis[dst]
// 128 channels -> 4 coalesced f32 atomics per lane. hW is L2-resident (51MB).
// ---------------------------------------------------------------------------
__global__ __launch_bounds__(256) void gcn_scatter(
    const float* __restrict__ hw, const float* __restrict__ dis,
    const int* __restrict__ srcI, const int* __restrict__ dstI,
    float* __restrict__ agg, int E)
{
    int gwave  = (blockIdx.x * blockDim.x + threadIdx.x) >> 5;
    int nwave  = (gridDim.x * blockDim.x) >> 5;
    int lane   = threadIdx.x & 31;
    for (int e = gwave; e < E; e += nwave) {
        int s = srcI[e];
        int d = dstI[e];
        float coef = dis[s] * dis[d];
        const float* hrow = hw  + (size_t)s * IN_CH;
        float*       arow = agg + (size_t)d * IN_CH;
        #pragma unroll
        for (int c = 0; c < IN_CH; c += 32)
            atomicAdd(&arow[c + lane], hrow[c + lane] * coef);
    }
}

// ---------------------------------------------------------------------------
// out = relu(agg + b)
// ---------------------------------------------------------------------------
__global__ void gcn_finalize(const float* __restrict__ agg, const float* __restrict__ b,
                             float* __restrict__ out, int total) {
    int idx = blockIdx.x * blockDim.x + threadIdx.x;
    if (idx < total)
        out[idx] = fmaxf(agg[idx] + b[idx & (IN_CH - 1)], 0.0f);
}

// ---------------------------------------------------------------------------
// Driver
// ---------------------------------------------------------------------------
extern "C" void kernel_launch(void* const* d_in, const int* in_sizes, int n_in,
                              void* d_out, int out_size, void* d_ws, size_t ws_size,
                              hipStream_t stream) {
    const float* x  = (const float*)d_in[0];
    const int*   ei = (const int*)d_in[1];     // [2, E] int32 (JAX canonicalized)
    const float* W1 = (const float*)d_in[2];
    const float* b1 = (const float*)d_in[3];
    const float* W2 = (const float*)d_in[4];
    const float* b2 = (const float*)d_in[5];
    float* out = (float*)d_out;

    const int N = in_sizes[0] / IN_CH;         // 100000
    const int E = in_sizes[1] / 2;             // 1600000
    const int* srcI = ei;                      // row 0: message sources
    const int* dstI = ei + E;                  // row 1: aggregation targets
    const int total = N * IN_CH;

    // Scratch layout in d_ws (floats): dis[N] | tmp[N*128] | agg[N*128]
    float* dis = (float*)d_ws;
    float* tmp = dis + (((size_t)N + 255) & ~(size_t)255);
    float* agg = tmp + (size_t)N * IN_CH;

    const int T = 256;
    const int nBlk = (N + T - 1) / T;
    const int tBlk = (total + T - 1) / T;
    const int gBlk = (N + MTILES * 16 - 1) / (MTILES * 16);  // 64-row strips

    // Normalization coefficients
    gcn_deg_init <<<nBlk, T, 0, stream>>>(dis, N);
    gcn_deg_accum<<<4096, T, 0, stream>>>(dstI, dis, E);
    gcn_deg_rsqrt<<<nBlk, T, 0, stream>>>(dis, N);

    // ---- Layer 1: h1 = relu(Ahat (x W1) + b1) -> d_out ----
    gcn_gemm_wmma<<<gBlk, T, 0, stream>>>(x, W1, tmp, N);
    gcn_agg_init <<<tBlk, T, 0, stream>>>(tmp, dis, agg, total);
    gcn_scatter  <<<16384, T, 0, stream>>>(tmp, dis, srcI, dstI, agg, E);
    gcn_finalize <<<tBlk, T, 0, stream>>>(agg, b1, out, total);

    // ---- Layer 2: out = relu(Ahat (h1 W2) + b2) -> d_out ----
    gcn_gemm_wmma<<<gBlk, T, 0, stream>>>(out, W2, tmp, N);
    gcn_agg_init <<<tBlk, T, 0, stream>>>(tmp, dis, agg, total);
    gcn_scatter  <<<16384, T, 0, stream>>>(tmp, dis, srcI, dstI, agg, E);
    gcn_finalize <<<tBlk, T, 0, stream>>>(agg, b2, out, total);
}